// CausalConv1d_90778428768580
// MI455X (gfx1250) — compile-verified
//
#include <hip/hip_runtime.h>
#include <hip/hip_bf16.h>

// Depthwise causal conv1d (K=16) as banded-Toeplitz WMMA matmuls on gfx1250.
//
// Roofline: 537 MB mandatory traffic -> >=23 us @ 23.3 TB/s. Memory-bound, so
// the 16-tap MACs ride V_WMMA_F32_16X16X4_F32 (8 per 256 outputs) and the
// triangular weight masks are realized by a zero-padded tap table in LDS
// (branch-free gathers), keeping the instruction stream load/store dominated.

typedef __attribute__((ext_vector_type(2))) float v2f;
typedef __attribute__((ext_vector_type(8))) float v8f;

#define DW_DIM   2048
#define DW_LEN   8192
#define DW_KW    16
#define DW_NB    4
#define DW_TILE  256                         // output positions per 16x16 D tile
#define DW_TPW   4                           // tiles per wave (B-frag reuse)
#define DW_WPC   (DW_LEN / (DW_TILE * DW_TPW))   // 8 waves per (n,c) row

// One 256-output tile: D[t][j] = sum_k Xprev[t][k]*Wprev[k][j] + Xcur[t][k]*Wcur[k][j]
template <bool GUARD>
__device__ __forceinline__ void dw_tile(const float* __restrict__ xb,
                                        float* __restrict__ ob,
                                        const v2f* bp, const v2f* bc,
                                        int t, int kh, float bv, int soff)
{
    v8f acc;
#pragma unroll
    for (int r = 0; r < 8; ++r) acc[r] = bv;     // bias seed

#pragma unroll
    for (int kk = 0; kk < 4; ++kk) {
        const int k0 = 4 * kk + kh;              // K index of VGPR0 (even)
        // A fragments: aligned float2 -> global_load_b64, offsets fold to IOFFSET
        v2f ac = *(const v2f*)(xb + 16 * t + k0);
        v2f ap;
        if (GUARD) {
            // First tile of the channel: row t==0 of the halo is the causal pad.
            // Unconditional load from a clamped in-range address, then cndmask.
            const bool pad = (t == 0);
            const int  po  = pad ? 0 : (16 * t + k0 - 16);
            ap = *(const v2f*)(xb + po);
            ap.x = pad ? 0.0f : ap.x;
            ap.y = pad ? 0.0f : ap.y;
        } else {
            ap = *(const v2f*)(xb + 16 * t + k0 - 16);
        }
        // Chained accumulate; whole wave executes (EXEC all-ones).
        acc = __builtin_amdgcn_wmma_f32_16x16x4_f32(
                  false, ap, false, bp[kk], (short)0, acc, false, false);
        acc = __builtin_amdgcn_wmma_f32_16x16x4_f32(
                  false, ac, false, bc[kk], (short)0, acc, false, false);
    }

    // D layout: VGPR r -> row r + 8*(lane>>4), col lane&15.
    // Each half-wave writes a contiguous 64B segment per r.
#pragma unroll
    for (int r = 0; r < 8; ++r) ob[soff + 16 * r] = acc[r];
}

__global__ __launch_bounds__(256) void
CausalConv1d_90778428768580_kernel(const float* __restrict__ x,
                                   const float* __restrict__ w,
                                   const float* __restrict__ bias,
                                   float* __restrict__ out)
{
    __shared__ float Etab[8][64];                // per-wave padded tap table

    const int lane = threadIdx.x & 31;
    const int wv   = threadIdx.x >> 5;

    // One wave per 1024 contiguous positions of one channel.
    const int gw = blockIdx.x * 8 + wv;          // 0 .. 65535
    const int cw = gw & (DW_WPC - 1);            // which eighth of the row
    const int nc = gw / DW_WPC;                  // n*DIM + c
    const int c  = nc & (DW_DIM - 1);
    const int l0 = cw * (DW_TILE * DW_TPW);
    const long base = (long)nc * DW_LEN + l0;

    // Zero-padded tap table: E[0..15]=0, E[16..31]=w[c][0..15], E[32..63]=0.
    // Then Wprev[k][j] = E[15+k-j], Wcur[k][j] = E[31+k-j] (triangles for free).
    float* E = Etab[wv];
    const float tap = w[c * DW_KW + (lane & 15)];
    E[lane]      = (lane < 16) ? 0.0f : tap;
    E[32 + lane] = 0.0f;

    // f32 WMMA 16x16x4 operand layouts (ISA 7.12.2):
    //   A (16x4): lanes 0-15 -> M=lane, VGPR0/1=K0/K1; lanes 16-31 -> K2/K3
    //   B (4x16): lanes 0-15 -> N=lane, VGPR0/1=K0/K1; lanes 16-31 -> K2/K3
    const int t  = lane & 15;                    // A row: 16-output group
    const int j  = lane & 15;                    // B col: position in group
    const int kh = (lane >> 4) * 2;              // K sub-offset {0,2}

    // Channel-invariant B fragments, reused across all DW_TPW tiles.
    v2f bp[4], bc[4];
#pragma unroll
    for (int kk = 0; kk < 4; ++kk) {
        const int i0 = 15 + 4 * kk + kh - j;     // in [0, 46]
        bp[kk].x = E[i0];      bp[kk].y = E[i0 + 1];
        bc[kk].x = E[i0 + 16]; bc[kk].y = E[i0 + 17];
    }

    const float bv  = bias[c];
    const int  soff = (lane >> 4) * 128 + (lane & 15);

    const float* xb = x + base;
    float*       ob = out + base;

    __builtin_prefetch(xb + DW_TILE, 0, 1);      // global_prefetch_b8 (speculative)
    if (__builtin_amdgcn_readfirstlane(l0) == 0) {
        dw_tile<true >(xb, ob, bp, bc, t, kh, bv, soff);   // causal-pad tile
    } else {
        dw_tile<false>(xb, ob, bp, bc, t, kh, bv, soff);
    }
#pragma unroll
    for (int tt = 1; tt < DW_TPW; ++tt) {
        __builtin_prefetch(xb + (tt + 1) * DW_TILE, 0, 1);
        dw_tile<false>(xb + tt * DW_TILE, ob + tt * DW_TILE,
                       bp, bc, t, kh, bv, soff);
    }
}

extern "C" void kernel_launch(void* const* d_in, const int* in_sizes, int n_in,
                              void* d_out, int out_size, void* d_ws, size_t ws_size,
                              hipStream_t stream) {
    (void)in_sizes; (void)n_in; (void)d_ws; (void)ws_size; (void)out_size;
    const float* x    = (const float*)d_in[0];
    const float* wgt  = (const float*)d_in[1];
    const float* bias = (const float*)d_in[2];
    float* out        = (float*)d_out;

    // 4*2048*8 = 65536 waves, 8 waves (256 threads) per block.
    const int total_waves = DW_NB * DW_DIM * DW_WPC;
    CausalConv1d_90778428768580_kernel<<<total_waves / 8, 256, 0, stream>>>(
        x, wgt, bias, out);
}